// AttentionEncoder_32650341384783
// MI455X (gfx1250) — compile-verified
//
#include <hip/hip_runtime.h>

typedef __attribute__((ext_vector_type(2))) float v2f;
typedef __attribute__((ext_vector_type(4))) float v4f;
typedef __attribute__((ext_vector_type(8))) float v8f;

#define HID 128   // hidden dim
#define KNB 32    // neighbors per node
#define NBS 132   // padded LDS row stride (floats): bank = (4*row + col) % 64 -> conflict-free

// Swap values between lane halves (lane L <-> lane L^16), wave32 group-of-32 swizzle:
// offset[15]=0, and_mask=0x1f, or_mask=0, xor_mask=0x10  => 0x401F (SWAPX16)
__device__ __forceinline__ float lane_swap16(float x) {
    int i = __float_as_int(x);
    i = __builtin_amdgcn_ds_swizzle(i, 0x401F);
    return __int_as_float(i);
}

__global__ __launch_bounds__(32) void
AttentionEncoder_node_kernel(const float* __restrict__ h_n,
                             const float* __restrict__ neighbor,
                             float* __restrict__ out,
                             int n_nodes)
{
    // Per-wave (== per-block) LDS: neighbor tile (padded), h vector, attn weights.
    __shared__ float lds_nb[KNB * NBS];   // 16896 B
    __shared__ float lds_h[HID];          //   512 B
    __shared__ float lds_attn[KNB];       //   128 B

    const int node = blockIdx.x;
    if (node >= n_nodes) return;          // wave-uniform; EXEC stays all-ones

    const int lane = threadIdx.x & 31;
    const int l15  = lane & 15;           // column / row-within-tile index
    const int half = lane >> 4;           // which K-pair of the WMMA fragment

    const float* hn = h_n      + (size_t)node * HID;
    const float* nb = neighbor + (size_t)node * (KNB * HID);

    // Stage h into LDS (32 lanes x float4 = 128 floats, fully coalesced).
    *(v4f*)&lds_h[lane * 4] = *(const v4f*)(hn + lane * 4);
    __syncthreads();

    // ---------------- Phase 1: scores = Nb(32x128) . h(128) via f32 WMMA ----------------
    // A fragment (16x4 fp32): lane L<16 -> row L, K={0,1}; lane L>=16 -> row L-16, K={2,3}.
    // B fragment (4x16 fp32): h chunk replicated across all 16 columns (broadcast LDS read).
    v8f acc0 = {0.f,0.f,0.f,0.f,0.f,0.f,0.f,0.f};  // k-rows 0..15
    v8f acc1 = {0.f,0.f,0.f,0.f,0.f,0.f,0.f,0.f};  // k-rows 16..31

    #pragma unroll
    for (int c = 0; c < 32; ++c) {
        const int dc = 4 * c + 2 * half;                       // d-offset for this lane's K pair
        v2f a0 = *(const v2f*)(nb + l15 * HID + dc);           // rows 0..15
        v2f a1 = *(const v2f*)(nb + (16 + l15) * HID + dc);    // rows 16..31
        // Stage the tile into LDS for phase 2 (each element written exactly once).
        *(v2f*)&lds_nb[l15 * NBS + dc]        = a0;
        *(v2f*)&lds_nb[(16 + l15) * NBS + dc] = a1;
        v2f b = *(const v2f*)&lds_h[dc];                       // broadcast, replicated columns
        acc0 = __builtin_amdgcn_wmma_f32_16x16x4_f32(false, a0, false, b, (short)0, acc0, false, false);
        acc1 = __builtin_amdgcn_wmma_f32_16x16x4_f32(false, a1, false, b, (short)0, acc1, false, false);
    }

    // ---------------- Softmax over 32 scores ----------------
    // D layout: VGPR j of acc0 holds score k = j + 8*half; acc1 holds k = 16 + j + 8*half
    // (value replicated across the 16 lanes of each half).
    const float scale = 0.08838834764831845f;  // 1/sqrt(128)
    float s[16];
    #pragma unroll
    for (int j = 0; j < 8; ++j) { s[j] = acc0[j] * scale; s[8 + j] = acc1[j] * scale; }

    float m = s[0];
    #pragma unroll
    for (int j = 1; j < 16; ++j) m = fmaxf(m, s[j]);
    m = fmaxf(m, lane_swap16(m));                 // global max across both halves

    float sum = 0.f;
    #pragma unroll
    for (int j = 0; j < 16; ++j) { s[j] = __expf(s[j] - m); sum += s[j]; }
    sum += lane_swap16(sum);                      // global sum
    const float inv = __builtin_amdgcn_rcpf(sum);

    // One lane per half publishes its 16 attention weights.
    if (l15 == 0) {
        #pragma unroll
        for (int j = 0; j < 8; ++j) {
            lds_attn[8 * half + j]      = s[j]     * inv;   // k = j + 8*half
            lds_attn[16 + 8 * half + j] = s[8 + j] * inv;   // k = 16 + j + 8*half
        }
    }
    __syncthreads();

    // ---------------- Phase 2: agg(1x128) = attn(1x32) . Nb(32x128) via f32 WMMA ----------------
    // A fragment: attn chunk replicated down all 16 rows -> every D row equals agg.
    v2f aA[8];
    #pragma unroll
    for (int c = 0; c < 8; ++c)
        aA[c] = *(const v2f*)&lds_attn[4 * c + 2 * half];   // broadcast ds_load_b64

    float* outn = out + (size_t)node * HID;
    #pragma unroll
    for (int t = 0; t < 8; ++t) {                 // 8 tiles of 16 output dims
        v8f acc = {0.f,0.f,0.f,0.f,0.f,0.f,0.f,0.f};
        #pragma unroll
        for (int c = 0; c < 8; ++c) {             // K = 32 in chunks of 4
            const int row = 4 * c + 2 * half;     // this lane's K rows of B
            v2f bB;
            bB.x = lds_nb[row * NBS       + 16 * t + l15];
            bB.y = lds_nb[(row + 1) * NBS + 16 * t + l15];
            acc = __builtin_amdgcn_wmma_f32_16x16x4_f32(false, aA[c], false, bB, (short)0, acc, false, false);
        }
        // Every row/VGPR of acc holds agg[16t + l15]; low half writes the residual-added result.
        if (half == 0) {
            outn[16 * t + l15] = lds_h[16 * t + l15] + acc[0];
        }
    }
}

extern "C" void kernel_launch(void* const* d_in, const int* in_sizes, int n_in,
                              void* d_out, int out_size, void* d_ws, size_t ws_size,
                              hipStream_t stream) {
    (void)n_in; (void)out_size; (void)d_ws; (void)ws_size;
    const float* h_n      = (const float*)d_in[0];   // (1, N, 128) fp32
    const float* neighbor = (const float*)d_in[1];   // (1, N, 32, 128) fp32
    float* out            = (float*)d_out;           // (1, N, 128) fp32

    const int n_nodes = in_sizes[0] / HID;           // 50000
    dim3 grid(n_nodes), block(32);                   // one wave32 per node
    hipLaunchKernelGGL(AttentionEncoder_node_kernel, grid, block, 0, stream,
                       h_n, neighbor, out, n_nodes);
}